// DGP_RFF_36249523978481
// MI455X (gfx1250) — compile-verified
//
#include <hip/hip_runtime.h>
#include <hip/hip_bf16.h>

// ---------------- problem constants (from reference) ----------------
constexpr int M_PTS  = 500000;   // points
constexpr int NSEG   = 25000;    // segments
constexpr int NRFF   = 256;      // random fourier features
constexpr int D0     = 64;       // input dim
constexpr int D1     = 128;      // hidden dim
constexpr int D2     = 64;       // output dim
constexpr long ZLEN  = 2L * NSEG * D2;          // sum_inv + sum_wm accumulators

// interleaved (WMMA-fragment-order) weight copies in workspace
constexpr int OM0LEN = D0  * NRFF;   // 16384
constexpr int WM0LEN = NRFF * D1;    // 32768
constexpr int E0LEN  = NRFF * D1;    // 32768
constexpr int OM1LEN = D1  * NRFF;   // 32768
constexpr int WM1LEN = NRFF * D2;    // 16384
constexpr int E1LEN  = NRFF * D2;    // 16384

typedef float v2f __attribute__((ext_vector_type(2)));
typedef float v8f __attribute__((ext_vector_type(8)));

#define LSTR 260   // padded LDS row stride (floats): 260 % 64 = 4 -> conflict-free A-frag reads

static __device__ __forceinline__ v8f wmma4(v2f a, v2f b, v8f c) {
    // V_WMMA_F32_16X16X4_F32 : D = A(16x4) * B(4x16) + C(16x16)
    return __builtin_amdgcn_wmma_f32_16x16x4_f32(
        /*neg_a=*/false, a, /*neg_b=*/false, b,
        /*c_mod=*/(short)0, c, /*reuse_a=*/false, /*reuse_b=*/false);
}

// interleave map: W[K][N] row-major  ->  Wt[(k>>1)*2N + n*2 + (k&1)]
// so a lane's B fragment (rows k0, k0+1 at column n) is one contiguous float2.
static __device__ __forceinline__ long ilv(int k, int n, int N) {
    return (long)(k >> 1) * (2 * N) + (n << 1) + (k & 1);
}

// -------------------------------------------------------------------
// Kernel 1: zero segment accumulators, build interleaved weights
//           (exp() folded into E0t/E1t)
// -------------------------------------------------------------------
__global__ void dgp_init_ws(const float* __restrict__ Om0, const float* __restrict__ Wm0,
                            const float* __restrict__ Wlv0,
                            const float* __restrict__ Om1, const float* __restrict__ Wm1,
                            const float* __restrict__ Wlv1,
                            float* __restrict__ ws) {
    long i = (long)blockIdx.x * blockDim.x + threadIdx.x;
    if (i < ZLEN) { ws[i] = 0.0f; return; }
    long j = i - ZLEN;
    float* Om0t = ws + ZLEN;
    float* Wm0t = Om0t + OM0LEN;
    float* E0t  = Wm0t + WM0LEN;
    float* Om1t = E0t  + E0LEN;
    float* Wm1t = Om1t + OM1LEN;
    float* E1t  = Wm1t + WM1LEN;
    if (j < OM0LEN) {                                   // Omega0 [64,256]
        int k = (int)(j >> 8), n = (int)(j & 255);
        Om0t[ilv(k, n, NRFF)] = Om0[j];
        return;
    }
    j -= OM0LEN;
    if (j < WM0LEN) {                                   // Wm0 [256,128]
        int k = (int)(j >> 7), n = (int)(j & 127);
        Wm0t[ilv(k, n, D1)] = Wm0[j];
        return;
    }
    j -= WM0LEN;
    if (j < E0LEN) {                                    // exp(Wlv0) [256,128]
        int k = (int)(j >> 7), n = (int)(j & 127);
        E0t[ilv(k, n, D1)] = __builtin_expf(Wlv0[j]);
        return;
    }
    j -= E0LEN;
    if (j < OM1LEN) {                                   // Omega1 [128,256]
        int k = (int)(j >> 8), n = (int)(j & 255);
        Om1t[ilv(k, n, NRFF)] = Om1[j];
        return;
    }
    j -= OM1LEN;
    if (j < WM1LEN) {                                   // Wm1 [256,64]
        int k = (int)(j >> 6), n = (int)(j & 63);
        Wm1t[ilv(k, n, D2)] = Wm1[j];
        return;
    }
    j -= WM1LEN;
    if (j < E1LEN) {                                    // exp(Wlv1) [256,64]
        int k = (int)(j >> 6), n = (int)(j & 63);
        E1t[ilv(k, n, D2)] = __builtin_expf(Wlv1[j]);
        return;
    }
}

// -------------------------------------------------------------------
// Kernel 2: fused 2-layer RFF-DGP pipeline, one wave32 per 16-row tile
// -------------------------------------------------------------------
__global__ __launch_bounds__(32)
void dgp_fused(const float* __restrict__ x,
               const float* __restrict__ Om0t, const float* __restrict__ b0v,
               const float* __restrict__ Wm0t,
               const float* __restrict__ Om1t, const float* __restrict__ b1v,
               const float* __restrict__ Wm1t,
               const float* __restrict__ E0t,  const float* __restrict__ E1t,
               const int*   __restrict__ xidx,
               float* __restrict__ sum_inv, float* __restrict__ sum_wm) {
    __shared__ float phiS[16 * LSTR];   // 16 x 256 intermediate (phi0, later phi1)
    __shared__ float mvS [16 * LSTR];   // x tile, later mean1 (cols 0..127) + var1 (cols 130..257)

    const int lane = threadIdx.x;       // wave32
    const int n16  = lane & 15;         // N index within 16-wide tile / M index for A frags
    const int kh   = lane >> 4;         // half-wave selects K pair
    const long m0  = (long)blockIdx.x * 16;
    const float C0 = 0.08838834764831845f;  // sqrt(2/256)

    // speculative prefetch of the next tile's x rows (global_prefetch_b8)
    if (m0 + 16 < M_PTS)
        __builtin_prefetch(&x[(m0 + 16) * D0 + lane * 8], 0, 0);

    // ---- stage x tile [16 x 64] into LDS (coalesced) ----
    for (int i = lane; i < 16 * D0; i += 32)
        mvS[(i >> 6) * LSTR + (i & 63)] = x[m0 * D0 + i];
    __syncthreads();

    // ---- GEMM1: z0 = x @ Omega0 + b0 ; phi0 = C0*cos(z0) -> phiS ----
    #pragma unroll 1
    for (int nt = 0; nt < 16; ++nt) {
        v8f acc = {0.f,0.f,0.f,0.f,0.f,0.f,0.f,0.f};
        const int col = nt * 16 + n16;
        #pragma unroll 4
        for (int kk = 0; kk < 16; ++kk) {
            const int k0 = 4 * kk + 2 * kh;
            v2f a; a.x = mvS[n16 * LSTR + k0]; a.y = mvS[n16 * LSTR + k0 + 1];
            v2f b = *(const v2f*)&Om0t[(long)(k0 >> 1) * (2 * NRFF) + (col << 1)];
            acc = wmma4(a, b, acc);
        }
        const float bb = b0v[col];
        #pragma unroll
        for (int r = 0; r < 8; ++r)
            phiS[(r + 8 * kh) * LSTR + col] = C0 * __builtin_cosf(acc[r] + bb);
    }
    __syncthreads();

    // ---- GEMM2/3: mean1 = phi0 @ Wm0 ; var1 = phi0^2 @ exp(Wlv0) -> mvS ----
    #pragma unroll 1
    for (int nt = 0; nt < 8; ++nt) {
        v8f am = {0.f,0.f,0.f,0.f,0.f,0.f,0.f,0.f};
        v8f av = {0.f,0.f,0.f,0.f,0.f,0.f,0.f,0.f};
        const int col = nt * 16 + n16;
        #pragma unroll 4
        for (int kk = 0; kk < 64; ++kk) {
            const int k0 = 4 * kk + 2 * kh;
            v2f a;  a.x = phiS[n16 * LSTR + k0]; a.y = phiS[n16 * LSTR + k0 + 1];
            v2f a2 = a * a;
            v2f bm = *(const v2f*)&Wm0t[(long)(k0 >> 1) * (2 * D1) + (col << 1)];
            v2f bv = *(const v2f*)&E0t [(long)(k0 >> 1) * (2 * D1) + (col << 1)];
            am = wmma4(a,  bm, am);
            av = wmma4(a2, bv, av);
        }
        #pragma unroll
        for (int r = 0; r < 8; ++r) {
            const int row = (r + 8 * kh) * LSTR;
            mvS[row + col]       = am[r];          // mean1 at cols 0..127
            mvS[row + 130 + col] = av[r];          // var1  at cols 130..257
        }
    }
    __syncthreads();

    // ---- GEMM4/5: z1 = mean1 @ Omega1 + b1 ; damp = exp(-0.5 * var1 @ Omega1^2)
    //      phi1 = C0*cos(z1)*damp -> phiS (overwrites phi0) ----
    #pragma unroll 1
    for (int nt = 0; nt < 16; ++nt) {
        v8f az = {0.f,0.f,0.f,0.f,0.f,0.f,0.f,0.f};
        v8f aw = {0.f,0.f,0.f,0.f,0.f,0.f,0.f,0.f};
        const int col = nt * 16 + n16;
        #pragma unroll 4
        for (int kk = 0; kk < 32; ++kk) {
            const int k0 = 4 * kk + 2 * kh;
            v2f amn; amn.x = mvS[n16 * LSTR + k0];       amn.y = mvS[n16 * LSTR + k0 + 1];
            v2f avr; avr.x = mvS[n16 * LSTR + 130 + k0]; avr.y = mvS[n16 * LSTR + 130 + k0 + 1];
            v2f bz = *(const v2f*)&Om1t[(long)(k0 >> 1) * (2 * NRFF) + (col << 1)];
            v2f bw = bz * bz;                            // (Omega1*Omega1) fragment, in-register
            az = wmma4(amn, bz, az);
            aw = wmma4(avr, bw, aw);
        }
        const float bb = b1v[col];
        #pragma unroll
        for (int r = 0; r < 8; ++r) {
            const float damp = __builtin_expf(-0.5f * aw[r]);
            phiS[(r + 8 * kh) * LSTR + col] = C0 * __builtin_cosf(az[r] + bb) * damp;
        }
    }
    __syncthreads();

    // ---- GEMM6/7: m2 = phi1 @ Wm1 ; v2 = phi1^2 @ exp(Wlv1) (kept in VGPRs) ----
    v8f m2[4], vv[4];
    #pragma unroll
    for (int nt = 0; nt < 4; ++nt) {
        v8f am = {0.f,0.f,0.f,0.f,0.f,0.f,0.f,0.f};
        v8f av = {0.f,0.f,0.f,0.f,0.f,0.f,0.f,0.f};
        const int col = nt * 16 + n16;
        #pragma unroll 4
        for (int kk = 0; kk < 64; ++kk) {
            const int k0 = 4 * kk + 2 * kh;
            v2f a;  a.x = phiS[n16 * LSTR + k0]; a.y = phiS[n16 * LSTR + k0 + 1];
            v2f a2 = a * a;
            v2f bm = *(const v2f*)&Wm1t[(long)(k0 >> 1) * (2 * D2) + (col << 1)];
            v2f bv = *(const v2f*)&E1t [(long)(k0 >> 1) * (2 * D2) + (col << 1)];
            am = wmma4(a,  bm, am);
            av = wmma4(a2, bv, av);
        }
        m2[nt] = am;
        vv[nt] = av;
    }

    // ---- epilogue: L2 row-normalize, inverse-variance weights, segment atomics ----
    // C-layout: VGPR r, half kh  <->  row m = r + 8*kh ; lanes of a half cover 16 cols
    float rinv[8];
    #pragma unroll
    for (int r = 0; r < 8; ++r) {
        float s = 0.0f;
        #pragma unroll
        for (int nt = 0; nt < 4; ++nt) { const float t = m2[nt][r]; s += t * t; }
        s += __shfl_xor(s, 1, 32);
        s += __shfl_xor(s, 2, 32);
        s += __shfl_xor(s, 4, 32);
        s += __shfl_xor(s, 8, 32);          // reduce within each 16-lane half
        rinv[r] = 1.0f / fmaxf(sqrtf(s), 1e-12f);
    }
    int seg[8];
    #pragma unroll
    for (int r = 0; r < 8; ++r)
        seg[r] = xidx[m0 + r + 8 * kh];

    #pragma unroll
    for (int nt = 0; nt < 4; ++nt) {
        const int col = nt * 16 + n16;
        #pragma unroll
        for (int r = 0; r < 8; ++r) {
            const float iv  = 1.0f / vv[nt][r];
            const float wmv = iv * m2[nt][r] * rinv[r];
            const long  o   = (long)seg[r] * D2 + col;
            atomicAdd(&sum_inv[o], iv);
            atomicAdd(&sum_wm[o],  wmv);
        }
    }
}

// -------------------------------------------------------------------
// Kernel 3: finalize  (out = [sum_wm/sum_inv , 1/sum_inv] concatenated)
// -------------------------------------------------------------------
__global__ void dgp_finalize(const float* __restrict__ sum_inv,
                             const float* __restrict__ sum_wm,
                             float* __restrict__ out) {
    long i = (long)blockIdx.x * blockDim.x + threadIdx.x;
    if (i < (long)NSEG * D2) {
        const float ev = 1.0f / sum_inv[i];
        out[i]                   = ev * sum_wm[i];
        out[(long)NSEG * D2 + i] = ev;
    }
}

// -------------------------------------------------------------------
extern "C" void kernel_launch(void* const* d_in, const int* in_sizes, int n_in,
                              void* d_out, int out_size, void* d_ws, size_t ws_size,
                              hipStream_t stream) {
    const float* x    = (const float*)d_in[0];
    const float* Om0  = (const float*)d_in[1];
    const float* b0   = (const float*)d_in[2];
    const float* Wm0  = (const float*)d_in[3];
    const float* Wlv0 = (const float*)d_in[4];
    const float* Om1  = (const float*)d_in[5];
    const float* b1   = (const float*)d_in[6];
    const float* Wm1  = (const float*)d_in[7];
    const float* Wlv1 = (const float*)d_in[8];
    const int*   xidx = (const int*)d_in[9];
    float* out = (float*)d_out;

    float* ws      = (float*)d_ws;
    float* sum_inv = ws;                       // [NSEG*64]
    float* sum_wm  = ws + (long)NSEG * D2;     // [NSEG*64]
    float* Om0t    = ws + ZLEN;                // interleaved weights
    float* Wm0t    = Om0t + OM0LEN;
    float* E0t     = Wm0t + WM0LEN;
    float* Om1t    = E0t  + E0LEN;
    float* Wm1t    = Om1t + OM1LEN;
    float* E1t     = Wm1t + WM1LEN;

    const long init_total = ZLEN + OM0LEN + WM0LEN + E0LEN + OM1LEN + WM1LEN + E1LEN;
    dgp_init_ws<<<(unsigned)((init_total + 255) / 256), 256, 0, stream>>>(
        Om0, Wm0, Wlv0, Om1, Wm1, Wlv1, ws);

    const int tiles = M_PTS / 16;              // 31250 wave-tiles
    dgp_fused<<<tiles, 32, 0, stream>>>(x, Om0t, b0, Wm0t, Om1t, b1, Wm1t,
                                        E0t, E1t, xidx, sum_inv, sum_wm);

    const long fin_total = (long)NSEG * D2;
    dgp_finalize<<<(unsigned)((fin_total + 255) / 256), 256, 0, stream>>>(sum_inv, sum_wm, out);
}